// MultiAssetHedgeModel_84189948936196
// MI455X (gfx1250) — compile-verified
//
#include <hip/hip_runtime.h>

// CDNA5 / gfx1250, wave32 only.
// Transposed-MLP scan using V_WMMA_F32_16X16X4_F32 (fp32, K=4 == IN_DIM).
//
// Per wave: TWO tiles of 16 batch rows (32 rows) across all 30 steps, so every
// per-step weight/bias load and the W2 zero-masking is amortized over 2 tiles.
//   layer1: h^T(32x16) = W1p(32x4) * x^T(4x16) + b1p    -> 2 WMMAs / tile
//   layer2: out^T(16x16) = W2pad(16x32) * h^T(32x16)+b2 -> 8 WMMAs / tile
//           (two independent accumulator chains per tile -> 4 chains total)
//
// KEY TRICK: W1's rows (hidden units) are consumed in permuted order
//     sigma: 0,1,4,5,8,9,12,13, 2,3,6,7,10,11,14,15   (per 16-row tile)
// chosen so the layer-1 D tiles (VGPR r = rows {sigma(r), sigma(r+8)}) are
// *identical* to the B-layout 4x16 K-chunks needed by layer 2:
//     B_chunk_c = (D[2c], D[2c+1])
// -> zero cross-lane shuffles between the layers. Only the 2-wide delta
// feedback needs a half-wave swap (ds_swizzle SWAPX16) per tile per step.

#define BATCH 524288
#define STEPS 30
#define HID   32

typedef float v2f __attribute__((ext_vector_type(2)));
typedef float v8f __attribute__((ext_vector_type(8)));

__device__ __forceinline__ float swap16(float x) {
    // ds_swizzle group-of-32: and=0x1f, or=0, xor=0x10 -> swap lanes 0-15 <-> 16-31
    return __int_as_float(__builtin_amdgcn_ds_swizzle(__float_as_int(x), 0x401f));
}

__global__ __launch_bounds__(256)
void hedge_scan_wmma(const float* __restrict__ market,  // (BATCH, STEPS+1, 2)
                     const float* __restrict__ W1,      // (STEPS, 32, 4)
                     const float* __restrict__ b1,      // (STEPS, 32)
                     const float* __restrict__ W2,      // (STEPS, 2, 32)
                     const float* __restrict__ b2,      // (STEPS, 2)
                     float* __restrict__ out)           // (BATCH, STEPS, 2)
{
    const int lane = threadIdx.x & 31;
    const int wave = threadIdx.x >> 5;
    const int b0   = (blockIdx.x * 8 + wave) * 32;   // 32 batch rows per wave
    const int half = lane >> 4;               // 0 = lanes 0-15, 1 = lanes 16-31
    const int l16  = lane & 15;
    const int l16c = (l16 < 2) ? l16 : 0;     // clamped W2 row (branchless A2 load)
    const bool avalid = (l16 < 2);

    // sigma(l16): permuted W1 row this lane feeds into the A tile (loop-invariant)
    const int sig = ((l16 < 8) ? (2 * l16) : (2 * l16 - 14)) - (l16 & 1);
    // float offset into a (32x4) W1 step block for A-layout 16x4 loads:
    //   V0/V1 hold K={0,1} for lanes 0-15 and K={2,3} for lanes 16-31
    const int a1off = sig * 4 + half * 2;

    // feedback deltas, valid in lanes 0-15 (tile p: batch row = b0 + 16p + lane)
    float d0[2] = {0.0f, 0.0f};
    float d1[2] = {0.0f, 0.0f};

    for (int t = 0; t < STEPS; ++t) {
        // ---------- B1 tiles = x^T (4x16) per batch tile:
        //   V0: lanes0-15 = K0 = m0,  lanes16-31 = K2 = delta0
        //   V1: lanes0-15 = K1 = m1,  lanes16-31 = K3 = delta1
        v2f bx[2];
        #pragma unroll
        for (int p = 0; p < 2; ++p) {
            const v2f m = *(const v2f*)(
                market + ((size_t)(b0 + 16 * p + l16) * (STEPS + 1) + t) * 2);
            const float s0 = swap16(d0[p]);
            const float s1 = swap16(d1[p]);
            bx[p].x = half ? s0 : m.x;
            bx[p].y = half ? s1 : m.y;
        }

        // ---------- A1 tiles (shared): W1 rows in sigma order
        const float* w1t = W1 + t * HID * 4;
        const v2f a1a = *(const v2f*)(w1t + a1off);        // rows sigma(l16)
        const v2f a1b = *(const v2f*)(w1t + 64 + a1off);   // rows 16+sigma(l16)

        // ---------- C = bias tiles (shared), sigma-permuted. Permuted indices
        // pair contiguously: lo-half lanes (0,1),(4,5),(8,9),(12,13);
        // hi-half lanes (2,3),(6,7),(10,11),(14,15)  -> 4x b64 per tile.
        const float* b1t = b1 + t * HID;
        v8f c1a, c1b;
        #pragma unroll
        for (int rr = 0; rr < 4; ++rr) {
            const v2f pa = *(const v2f*)(b1t + rr * 4 + half * 2);
            const v2f pb = *(const v2f*)(b1t + 16 + rr * 4 + half * 2);
            c1a[2 * rr]     = pa.x;
            c1a[2 * rr + 1] = pa.y;
            c1b[2 * rr]     = pb.x;
            c1b[2 * rr + 1] = pb.y;
        }

        // ---------- layer 1: 4 independent 16x16x4 fp32 WMMAs
        v8f h_lo[2], h_hi[2];
        #pragma unroll
        for (int p = 0; p < 2; ++p) {
            h_lo[p] = __builtin_amdgcn_wmma_f32_16x16x4_f32(
                false, a1a, false, bx[p], (short)0, c1a, false, false);
            h_hi[p] = __builtin_amdgcn_wmma_f32_16x16x4_f32(
                false, a1b, false, bx[p], (short)0, c1b, false, false);
        }

        // relu (dual-issues as v_dual_max_num pairs)
        #pragma unroll
        for (int p = 0; p < 2; ++p) {
            #pragma unroll
            for (int r = 0; r < 8; ++r) {
                h_lo[p][r] = fmaxf(h_lo[p][r], 0.0f);
                h_hi[p][r] = fmaxf(h_hi[p][r], 0.0f);
            }
        }

        // ---------- layer 2: out^T = W2pad * h^T + b2, eight 16x16x4 K-chunks.
        // Thanks to the sigma permutation, B chunk c is just (D[2c], D[2c+1]).
        const float* w2t = W2 + t * 2 * HID;
        const float* b2t = b2 + t * 2;
        const float bb0 = b2t[0];   // uniform loads, no exec branches
        const float bb1 = b2t[1];
        v8f acc_a[2], acc_b[2];
        #pragma unroll
        for (int p = 0; p < 2; ++p) {
            acc_a[p] = (v8f){};
            acc_b[p] = (v8f){};
            acc_a[p][0] = half ? 0.0f : bb0;   // row M=0 -> out0 (M=8 pad = 0)
            acc_a[p][1] = half ? 0.0f : bb1;   // row M=1 -> out1
        }

        #pragma unroll
        for (int c = 0; c < 8; ++c) {
            // A chunk (shared): W2 columns 4c..4c+3, rows 0..15 (rows 0,1 real).
            // Branchless: every lane loads a valid (clamped) address, zeros after.
            const v2f a2r = *(const v2f*)(w2t + l16c * HID + c * 4 + half * 2);
            v2f a2;
            a2.x = avalid ? a2r.x : 0.0f;
            a2.y = avalid ? a2r.y : 0.0f;

            const int u = 2 * (c & 3);
            #pragma unroll
            for (int p = 0; p < 2; ++p) {
                const v8f& D = (c & 4) ? h_hi[p] : h_lo[p];
                v2f bh;
                bh.x = D[u];
                bh.y = D[u + 1];
                if (c & 1) {
                    acc_b[p] = __builtin_amdgcn_wmma_f32_16x16x4_f32(
                        false, a2, false, bh, (short)0, acc_b[p], false, false);
                } else {
                    acc_a[p] = __builtin_amdgcn_wmma_f32_16x16x4_f32(
                        false, a2, false, bh, (short)0, acc_a[p], false, false);
                }
            }
        }

        // ---------- store deltas (NT: written once, never re-read), recycle
        #pragma unroll
        for (int p = 0; p < 2; ++p) {
            d0[p] = acc_a[p][0] + acc_b[p][0];   // lanes 0-15: out0
            d1[p] = acc_a[p][1] + acc_b[p][1];   // lanes 0-15: out1
        }
        if (!half) {
            #pragma unroll
            for (int p = 0; p < 2; ++p) {
                v2f o;
                o.x = d0[p];
                o.y = d1[p];
                __builtin_nontemporal_store(
                    o, (v2f*)(out + ((size_t)(b0 + 16 * p + l16) * STEPS + t) * 2));
            }
        }
    }
}

extern "C" void kernel_launch(void* const* d_in, const int* in_sizes, int n_in,
                              void* d_out, int out_size, void* d_ws, size_t ws_size,
                              hipStream_t stream) {
    (void)in_sizes; (void)n_in; (void)d_ws; (void)ws_size; (void)out_size;
    const float* market = (const float*)d_in[0];
    const float* W1     = (const float*)d_in[1];
    const float* b1     = (const float*)d_in[2];
    const float* W2     = (const float*)d_in[3];
    const float* b2     = (const float*)d_in[4];
    float*       out    = (float*)d_out;

    // 32 batch rows per wave, 8 waves per 256-thread block
    const int blocks = BATCH / (32 * 8);   // 2048
    hedge_scan_wmma<<<blocks, 256, 0, stream>>>(market, W1, b1, W2, b2, out);
}